// Net_82489141887386
// MI455X (gfx1250) — compile-verified
//
#include <hip/hip_runtime.h>

// ---------------------------------------------------------------------------
// GCN layer: xw = x@W ; agg = segment_sum(xw[src], dst) ; h = relu(agg + b)
// out = (sum_n h) @ Wd + bd
// Memory-bound (~155 MB HBM traffic -> ~6.6us floor @ 23.3 TB/s).
// GEMM uses V_WMMA_F32_16X16X4_F32, B fragments served branch-free from LDS.
// ---------------------------------------------------------------------------

typedef __attribute__((ext_vector_type(2))) float v2f;
typedef __attribute__((ext_vector_type(4))) float v4f;
typedef __attribute__((ext_vector_type(8))) float v8f;

#define N_NODES 100000
#define N_EDGES 6400000
#define F_IN    256
#define HIDDEN  8
#define N_TILES (N_NODES / 16)     // 6250 = 5 * 1250, exact
#define WAVES_PER_BLOCK 5          // 160 threads; 5 tiles/block; 1250 blocks
#define GEMM_BLOCK (WAVES_PER_BLOCK * 32)
#define GEMM_GRID  (N_TILES / WAVES_PER_BLOCK)

// ---------------- kernel 0: zero scratch (agg + pooled) ----------------
__global__ void zero_f32(float* __restrict__ p, int n) {
  int i = blockIdx.x * blockDim.x + threadIdx.x;
  if (i < n) p[i] = 0.0f;
}

// ---------------- kernel 1: xw = x @ W via WMMA f32 16x16x4 ----------------
// One wave computes a 16x16 tile of (x @ Wpad), Wpad = [W | 0] (16 cols).
// A frag (16x4 f32, 2 VGPRs): lane L holds A[M = L&15][K = v + 2*(L>>4)].
// B frag (4x16 f32, 2 VGPRs): lane L holds B[K = v + 2*(L>>4)][N = L&15].
// C/D  (16x16 f32, 8 VGPRs):  lane L holds D[M = v + 8*(L>>4)][N = L&15].
// B fragments come from LDS-resident transposed+padded W: Wt[16][256],
// so each lane's (b0,b1) is one contiguous 8-byte ds_load — no branches,
// EXEC stays all-1s through the whole WMMA loop.
__global__ __launch_bounds__(GEMM_BLOCK) void gemm_xw(
    const float* __restrict__ x,
    const float* __restrict__ W,
    float* __restrict__ xw) {
  __shared__ float Wt[16 * F_IN];           // 16 KB: Wt[n][k], cols >=8 zero

  // Cooperative stage: 4096 elements across 160 threads.
  for (int i = threadIdx.x; i < 16 * F_IN; i += GEMM_BLOCK) {
    const int n = i >> 8;                   // 0..15
    const int k = i & (F_IN - 1);           // 0..255
    Wt[i] = (n < HIDDEN) ? W[k * HIDDEN + n] : 0.0f;
  }
  __syncthreads();

  const int wave = threadIdx.x >> 5;        // 0..4
  const int lane = threadIdx.x & 31;
  const int tile = blockIdx.x * WAVES_PER_BLOCK + wave;   // always < N_TILES
  const int row0 = tile * 16;

  const int m  = lane & 15;                 // A row within tile / B,D column
  const int kh = (lane >> 4) * 2;           // K sub-offset: 0 or 2
  const float* xrow = x + (size_t)(row0 + m) * F_IN;
  const float* wrow = &Wt[m * F_IN];        // this lane's B column, contiguous

  v8f acc = {};
  #pragma unroll 8
  for (int k = 0; k < F_IN; k += 4) {
    v2f a  = *(const v2f*)(xrow + k + kh);  // global_load_b64
    v2f bf = *(const v2f*)(wrow + k + kh);  // ds_load_b64
    acc = __builtin_amdgcn_wmma_f32_16x16x4_f32(
        /*neg_a=*/false, a, /*neg_b=*/false, bf,
        /*c_mod=*/(short)0, acc, /*reuse_a=*/false, /*reuse_b=*/false);
  }

  // Write the 8 real columns. Lane L holds rows v + 8*(L>>4), column m.
  if (m < HIDDEN) {
    const int mhalf = (lane >> 4) * 8;
    #pragma unroll
    for (int v = 0; v < 8; ++v) {
      xw[(size_t)(row0 + v + mhalf) * HIDDEN + m] = acc[v];
    }
  }
}

// ---------------- kernel 2: agg[dst] += xw[src] (thread per edge) ----------
__global__ __launch_bounds__(256) void edge_scatter(const int* __restrict__ esrc,
                                                    const int* __restrict__ edst,
                                                    const float* __restrict__ xw,
                                                    float* __restrict__ agg) {
  int e = blockIdx.x * blockDim.x + threadIdx.x;
  if (e >= N_EDGES) return;
  const int s = esrc[e];
  const int d = edst[e];
  const v4f* sp = (const v4f*)(xw + (size_t)s * HIDDEN);  // 32B-aligned rows
  v4f m0 = sp[0];
  v4f m1 = sp[1];
  float* dp = agg + (size_t)d * HIDDEN;
  atomicAdd(dp + 0, m0.x); atomicAdd(dp + 1, m0.y);
  atomicAdd(dp + 2, m0.z); atomicAdd(dp + 3, m0.w);
  atomicAdd(dp + 4, m1.x); atomicAdd(dp + 5, m1.y);
  atomicAdd(dp + 6, m1.z); atomicAdd(dp + 7, m1.w);
}

// ---------------- kernel 3: pooled = sum_n relu(agg[n] + b) ----------------
__global__ __launch_bounds__(256) void reduce_pool(const float* __restrict__ agg,
                                                   const float* __restrict__ bias,
                                                   float* __restrict__ pooled) {
  float bv[HIDDEN];
  #pragma unroll
  for (int c = 0; c < HIDDEN; ++c) bv[c] = bias[c];

  float s[HIDDEN];
  #pragma unroll
  for (int c = 0; c < HIDDEN; ++c) s[c] = 0.0f;

  const int stride = gridDim.x * blockDim.x;
  for (int i = blockIdx.x * blockDim.x + threadIdx.x; i < N_NODES; i += stride) {
    const v4f* p = (const v4f*)(agg + (size_t)i * HIDDEN);
    v4f a0 = p[0];
    v4f a1 = p[1];
    s[0] += fmaxf(a0.x + bv[0], 0.0f);
    s[1] += fmaxf(a0.y + bv[1], 0.0f);
    s[2] += fmaxf(a0.z + bv[2], 0.0f);
    s[3] += fmaxf(a0.w + bv[3], 0.0f);
    s[4] += fmaxf(a1.x + bv[4], 0.0f);
    s[5] += fmaxf(a1.y + bv[5], 0.0f);
    s[6] += fmaxf(a1.z + bv[6], 0.0f);
    s[7] += fmaxf(a1.w + bv[7], 0.0f);
  }

  // wave32 reduction
  #pragma unroll
  for (int off = 16; off >= 1; off >>= 1) {
    #pragma unroll
    for (int c = 0; c < HIDDEN; ++c) s[c] += __shfl_down(s[c], off, 32);
  }
  if ((threadIdx.x & 31) == 0) {
    #pragma unroll
    for (int c = 0; c < HIDDEN; ++c) atomicAdd(&pooled[c], s[c]);
  }
}

// ---------------- kernel 4: out = pooled @ Wd + bd -------------------------
__global__ void final_dense(const float* __restrict__ pooled,
                            const float* __restrict__ Wd,
                            const float* __restrict__ bd,
                            float* __restrict__ out) {
  if (threadIdx.x == 0 && blockIdx.x == 0) {
    float acc = bd[0];
    #pragma unroll
    for (int c = 0; c < HIDDEN; ++c) acc += pooled[c] * Wd[c];
    out[0] = acc;
  }
}

extern "C" void kernel_launch(void* const* d_in, const int* in_sizes, int n_in,
                              void* d_out, int out_size, void* d_ws, size_t ws_size,
                              hipStream_t stream) {
  const float* x    = (const float*)d_in[0];   // [N, 256]
  const int*   esrc = (const int*)  d_in[1];   // [E]
  const int*   edst = (const int*)  d_in[2];   // [E]
  const float* W    = (const float*)d_in[3];   // [256, 8]
  const float* b    = (const float*)d_in[4];   // [8]
  const float* Wd   = (const float*)d_in[5];   // [8, 1]
  const float* bd   = (const float*)d_in[6];   // [1]
  float* out = (float*)d_out;

  // Workspace layout: xw[N*8] | agg[N*8] | pooled[8]  (~6.4 MB)
  float* xw     = (float*)d_ws;
  float* agg    = xw  + (size_t)N_NODES * HIDDEN;
  float* pooled = agg + (size_t)N_NODES * HIDDEN;

  // Zero agg + pooled (contiguous) every call — graph-replay safe.
  const int nz = N_NODES * HIDDEN + HIDDEN;
  zero_f32<<<(nz + 255) / 256, 256, 0, stream>>>(agg, nz);

  // xw = x @ W  (WMMA; 5 wave32 tiles per 160-thread block, exact grid)
  gemm_xw<<<GEMM_GRID, GEMM_BLOCK, 0, stream>>>(x, W, xw);

  // agg[dst] += xw[src]
  edge_scatter<<<(N_EDGES + 255) / 256, 256, 0, stream>>>(esrc, edst, xw, agg);

  // pooled = sum relu(agg + b)
  reduce_pool<<<200, 256, 0, stream>>>(agg, b, pooled);

  // out = pooled @ Wd + bd
  final_dense<<<1, 32, 0, stream>>>(pooled, Wd, bd, out);
}